// cAttend_simple_2911987827482
// MI455X (gfx1250) — compile-verified
//
#include <hip/hip_runtime.h>

typedef __attribute__((ext_vector_type(2))) float v2f;
typedef __attribute__((ext_vector_type(8))) float v8f;

#define NB   64      // batch
#define ND   8192    // dense dim
#define NE   256     // embed dim == qk dim
#define KSPLIT 16
#define KCHUNK (ND / KSPLIT)   // 512

__device__ __forceinline__ v8f wmma4(v2f a, v2f b, v8f c) {
  // V_WMMA_F32_16X16X4_F32 : D = A(16x4) * B(4x16) + C, full f32 accumulation
  return __builtin_amdgcn_wmma_f32_16x16x4_f32(false, a, false, b, (short)0, c,
                                               false, false);
}

// ---------------------------------------------------------------- reductions
__global__ void k_sv(const float* __restrict__ X, float* __restrict__ sv) {
  __shared__ float red[256];
  const int b = blockIdx.x;
  float s = 0.f;
  for (int d = threadIdx.x; d < ND; d += 256) s += X[b * ND + d];
  red[threadIdx.x] = s;
  __syncthreads();
  for (int off = 128; off > 0; off >>= 1) {
    if (threadIdx.x < off) red[threadIdx.x] += red[threadIdx.x + off];
    __syncthreads();
  }
  if (threadIdx.x == 0) sv[b] = red[0];
}

__global__ void k_c(const float* __restrict__ QV, const float* __restrict__ bk,
                    float* __restrict__ c) {
  __shared__ float red[256];
  const int b = blockIdx.x;
  red[threadIdx.x] = QV[b * NE + threadIdx.x] * bk[threadIdx.x];
  __syncthreads();
  for (int off = 128; off > 0; off >>= 1) {
    if (threadIdx.x < off) red[threadIdx.x] += red[threadIdx.x + off];
    __syncthreads();
  }
  if (threadIdx.x == 0) c[b] = red[0];
}

// ------------------------- partial EV: EVp[kc][64,256] = X[:,kc] @ E[1+kc,:]
// C[m,n] = sum_{k in chunk} X[m,k] * E[k+1,n]   (K split 16x for parallelism)
__global__ void k_gemm_ev_part(const float* __restrict__ X,
                               const float* __restrict__ E,
                               float* __restrict__ EVp) {
  const int lane = threadIdx.x & 31;
  const int wave = threadIdx.x >> 5;
  const int tile = blockIdx.x * 8 + wave;   // 1024 tiles: 16 (K) x 4 (M) x 16 (N)
  const int kc = tile >> 6;                 // k-chunk 0..15
  const int t  = tile & 63;
  const int m0 = (t >> 4) * 16;
  const int n0 = (t & 15) * 16;
  const int i  = lane & 15;
  const int kh = (lane >> 4) * 2;           // 0 | 2
  const int kbase = kc * KCHUNK;
  v8f acc = {};
#pragma unroll 4
  for (int kk = 0; kk < KCHUNK; kk += 4) {
    const int k0 = kbase + kk;
    v2f a, b;
    const float* ap = X + (m0 + i) * ND + k0 + kh;
    a.x = ap[0];
    a.y = ap[1];
    b.x = E[(k0 + kh + 1) * NE + n0 + i];
    b.y = E[(k0 + kh + 2) * NE + n0 + i];
    acc = wmma4(a, b, acc);
  }
  const int rb = m0 + (lane >> 4) * 8;
  const int col = n0 + i;
  float* out = EVp + kc * (NB * NE);
#pragma unroll
  for (int r = 0; r < 8; ++r) out[(rb + r) * NE + col] = acc[r];
}

// Deterministic fixed-order reduction of the 16 K-chunk partials.
__global__ void k_ev_reduce(const float* __restrict__ EVp, float* __restrict__ EV) {
  const int idx = blockIdx.x * 256 + threadIdx.x;   // 64 blocks -> 16384 elems
  float s = 0.f;
#pragma unroll
  for (int j = 0; j < KSPLIT; ++j) s += EVp[j * (NB * NE) + idx];
  EV[idx] = s;
}

// --------------------------------- QV[64,256] = EV @ Wq^T + sv (x) bq
// C[m,n] = sum_k EV[m,k] * Wq[n,k]  (+ sv[m]*bq[n])
__global__ void k_gemm_qv(const float* __restrict__ EV, const float* __restrict__ Wq,
                          const float* __restrict__ bq, const float* __restrict__ sv,
                          float* __restrict__ QV) {
  const int lane = threadIdx.x & 31;
  const int wave = threadIdx.x >> 5;
  const int tile = blockIdx.x * 8 + wave;   // 64 tiles
  const int m0 = (tile >> 4) * 16;
  const int n0 = (tile & 15) * 16;
  const int i  = lane & 15;
  const int kh = (lane >> 4) * 2;
  v8f acc = {};
#pragma unroll 8
  for (int k0 = 0; k0 < NE; k0 += 4) {
    v2f a, b;
    const float* ap = EV + (m0 + i) * NE + k0 + kh;
    a.x = ap[0];
    a.y = ap[1];
    const float* bp = Wq + (n0 + i) * NE + k0 + kh;
    b.x = bp[0];
    b.y = bp[1];
    acc = wmma4(a, b, acc);
  }
  const int rb = m0 + (lane >> 4) * 8;
  const int col = n0 + i;
#pragma unroll
  for (int r = 0; r < 8; ++r) {
    const int row = rb + r;
    QV[row * NE + col] = acc[r] + sv[row] * bq[col];
  }
}

// --------------------------------------------- W[64,256] = QV @ Wk
// C[m,n] = sum_k QV[m,k] * Wk[k,n]
__global__ void k_gemm_w(const float* __restrict__ QV, const float* __restrict__ Wk,
                         float* __restrict__ W) {
  const int lane = threadIdx.x & 31;
  const int wave = threadIdx.x >> 5;
  const int tile = blockIdx.x * 8 + wave;   // 64 tiles
  const int m0 = (tile >> 4) * 16;
  const int n0 = (tile & 15) * 16;
  const int i  = lane & 15;
  const int kh = (lane >> 4) * 2;
  v8f acc = {};
#pragma unroll 8
  for (int k0 = 0; k0 < NE; k0 += 4) {
    v2f a, b;
    const float* ap = QV + (m0 + i) * NE + k0 + kh;
    a.x = ap[0];
    a.y = ap[1];
    b.x = Wk[(k0 + kh) * NE + n0 + i];
    b.y = Wk[(k0 + kh + 1) * NE + n0 + i];
    acc = wmma4(a, b, acc);
  }
  const int rb = m0 + (lane >> 4) * 8;
  const int col = n0 + i;
#pragma unroll
  for (int r = 0; r < 8; ++r) W[(rb + r) * NE + col] = acc[r];
}

// ------------- Y[64,8192]: FX = W @ E[1:,:]^T ; y = x ? x*(1+scale*(FX+c)) : 0
// C[m,n] = sum_k W[m,k] * E[n+1,k]
__global__ void k_gemm_fx(const float* __restrict__ W, const float* __restrict__ E,
                          const float* __restrict__ X, const float* __restrict__ c,
                          float* __restrict__ Y) {
  const int lane = threadIdx.x & 31;
  const int wave = threadIdx.x >> 5;
  const int tile = blockIdx.x * 8 + wave;   // 2048 tiles: 4 (M) x 512 (N)
  const int m0 = (tile >> 9) * 16;
  const int n0 = (tile & 511) * 16;
  const int i  = lane & 15;
  const int kh = (lane >> 4) * 2;
  v8f acc = {};
#pragma unroll 8
  for (int k0 = 0; k0 < NE; k0 += 4) {
    v2f a, b;
    const float* ap = W + (m0 + i) * NE + k0 + kh;
    a.x = ap[0];
    a.y = ap[1];
    const float* bp = E + (n0 + i + 1) * NE + k0 + kh;
    b.x = bp[0];
    b.y = bp[1];
    acc = wmma4(a, b, acc);
  }
  const float scale = 0.0625f;  // 1/sqrt(256)
  const int rb = m0 + (lane >> 4) * 8;
  const int col = n0 + i;
#pragma unroll
  for (int r = 0; r < 8; ++r) {
    const int row = rb + r;
    const float xv = X[row * ND + col];
    const float fx = scale * (acc[r] + c[row]);
    Y[row * ND + col] = (xv == 0.0f) ? 0.0f : xv * (1.0f + fx);
  }
}

extern "C" void kernel_launch(void* const* d_in, const int* in_sizes, int n_in,
                              void* d_out, int out_size, void* d_ws, size_t ws_size,
                              hipStream_t stream) {
  (void)in_sizes; (void)n_in; (void)out_size; (void)ws_size;
  const float* X  = (const float*)d_in[1];  // [64, 8192]
  const float* E  = (const float*)d_in[2];  // [8193, 256]
  const float* Wq = (const float*)d_in[3];  // [256, 256]
  const float* bq = (const float*)d_in[4];  // [256]
  const float* Wk = (const float*)d_in[5];  // [256, 256]
  const float* bk = (const float*)d_in[6];  // [256]
  float* Y = (float*)d_out;                 // [64, 8192]

  float* ws  = (float*)d_ws;
  float* sv  = ws;                    // 64
  float* cc  = ws + 64;               // 64
  float* EV  = ws + 128;              // 64*256
  float* QV  = EV + NB * NE;          // 64*256
  float* W   = QV + NB * NE;          // 64*256
  float* EVp = W + NB * NE;           // 16 * 64*256  (~1 MB)

  k_sv          <<<NB,  256, 0, stream>>>(X, sv);
  k_gemm_ev_part<<<128, 256, 0, stream>>>(X, E, EVp);
  k_ev_reduce   <<<64,  256, 0, stream>>>(EVp, EV);
  k_gemm_qv     <<<8,   256, 0, stream>>>(EV, Wq, bq, sv, QV);
  k_c           <<<NB,  256, 0, stream>>>(QV, bk, cc);
  k_gemm_w      <<<8,   256, 0, stream>>>(QV, Wk, W);
  k_gemm_fx     <<<256, 256, 0, stream>>>(W, E, X, cc, Y);
}